// MeKooMatching_90924457656589
// MI455X (gfx1250) — compile-verified
//
#include <hip/hip_runtime.h>

typedef __attribute__((ext_vector_type(2))) float        v2f;
typedef __attribute__((ext_vector_type(4))) float        v4f;
typedef __attribute__((ext_vector_type(8))) float        v8f;
typedef __attribute__((ext_vector_type(4))) unsigned int v4u;
typedef __attribute__((ext_vector_type(4))) int          v4i;
typedef __attribute__((ext_vector_type(8))) int          v8i;

#define B_     8
#define T_     2048
#define D_     128
#define NK_    16
#define ROWS   (B_ * (T_ - 1))        // 16376 rows of seq1/seq2
#define NTILES ((ROWS + 15) / 16)     // 1024 row tiles
#define WPB    4                      // waves per block (4 row tiles / block)
#define LDSW   136                    // padded LDS row stride (dwords): +272dw for k+2 => +16 banks, conflict-free

// Block = 4 waves, one expert, 4 consecutive 16-row tiles. B_n (128x128 f32) is
// staged once per block into LDS via the CDNA5 Tensor Data Mover (padded rows),
// then each wave runs the fp32 WMMA pipeline V_WMMA_F32_16X16X4_F32 with B
// fragments served from the DS pipe, and accumulates sum(exp(|C - seq2|)).
__global__ __launch_bounds__(32 * WPB)
void mekoo_main(const float* __restrict__ seq,
                const float* __restrict__ Kp,
                float* __restrict__ per_expert) {
  __shared__ float ldsB[D_ * LDSW];   // 68 KB

  const int n     = blockIdx.y;
  const int wave  = threadIdx.x >> 5;
  const int lane  = threadIdx.x & 31;
  const int tile  = blockIdx.x * WPB + wave;
  const int lhalf = lane >> 4;
  const int l16   = lane & 15;
  const int khi   = lhalf * 2;

  const float* Bn = Kp + (size_t)n * D_ * D_;

  // ---- stage B_n into LDS (row stride LDSW dwords) ----
#if __has_builtin(__builtin_amdgcn_tensor_load_to_lds)
  if (wave == 0) {
    const unsigned ldsOff = (unsigned)(uintptr_t)&ldsB[0];  // generic addr low 32 = LDS byte offset
    const unsigned long long ga = (unsigned long long)(uintptr_t)Bn;
    v4u g0;
    g0.x = 1u;                                   // count=1, no gather, user descriptor
    g0.y = ldsOff;                               // lds_addr (bytes)
    g0.z = (unsigned)ga;                         // global_addr[31:0]
    g0.w = (unsigned)((ga >> 32) & 0x1FFFFFFu)   // global_addr[56:32]
         | (2u << 30);                           // type = 2 ("image")
    v8i g1;
    g1[0] = (int)((2u << 16)      // data_size = 4 bytes
                | (1u << 20)      // pad_enable
                | (6u << 22)      // pad_interval: 128 dwords
                | (7u << 25));    // pad_amount: 8 dwords -> LDS row stride 136 dw
    g1[1] = (int)(128u << 16);    // tensor_dim0 = 128 (low half in bits[79:64]=0)
    g1[2] = (int)(128u << 16);    // tensor_dim1 = 128
    g1[3] = (int)(128u << 16);    // tile_dim0 = 128
    g1[4] = 128;                  // tile_dim1 = 128, tile_dim2 = 0
    g1[5] = 128;                  // tensor_dim0_stride = 128 elements
    g1[6] = 0;                    // stride hi / tensor_dim1_stride lo
    g1[7] = 0;
    v4i gz = {0, 0, 0, 0};
#if __clang_major__ >= 23
    v8i gz8 = {0, 0, 0, 0, 0, 0, 0, 0};
    __builtin_amdgcn_tensor_load_to_lds(g0, g1, gz, gz, gz8, 0);
#else
    __builtin_amdgcn_tensor_load_to_lds(g0, g1, gz, gz, 0);
#endif
    __builtin_amdgcn_s_wait_tensorcnt(0);
  }
#else
  // fallback: cooperative float4 staging with the same padded layout
  for (int it = 0; it < 32; ++it) {
    const int idx = it * (32 * WPB) + threadIdx.x;  // 4096 float4 total
    const int row = idx >> 5;
    const int c4  = (idx & 31) << 2;
    *(v4f*)&ldsB[row * LDSW + c4] = *(const v4f*)&Bn[row * D_ + c4];
  }
#endif
  __syncthreads();

  // ---- A fragments: lane l16 owns row M=l16; VGPR g holds K = 4k + khi + g ----
  v2f a[32];
  {
    const int rowA = tile * 16 + l16;
    if (rowA < ROWS) {
      const int b = rowA / (T_ - 1);
      const int t = rowA - b * (T_ - 1);
      const float* arow = seq + (size_t)(b * T_ + t) * D_;
#pragma unroll
      for (int k = 0; k < 32; ++k) {
        const int col = 4 * k + khi;
        a[k].x = arow[col];
        a[k].y = arow[col + 1];
      }
    } else {
#pragma unroll
      for (int k = 0; k < 32; ++k) { a[k].x = 0.0f; a[k].y = 0.0f; }
    }
  }

  // ---- C-row mapping for epilogue: VGPR g holds row M = g + 8*lhalf ----
  const float* s2row[8];
  bool valid[8];
#pragma unroll
  for (int g = 0; g < 8; ++g) {
    const int r = tile * 16 + g + 8 * lhalf;
    valid[g] = (r < ROWS);
    const int rc = valid[g] ? r : 0;
    const int b  = rc / (T_ - 1);
    const int t  = rc - b * (T_ - 1);
    s2row[g] = seq + (size_t)(b * T_ + t + 1) * D_;   // seq2 row
  }

  float acc = 0.0f;
  for (int j = 0; j < 8; ++j) {       // 8 column tiles of 16
    v8f c = {};
#pragma unroll
    for (int k = 0; k < 32; ++k) {    // K = 128 in steps of 4
      const int kb = 4 * k + khi;     // B frag: lane l16 = col N, VGPR g = K kb+g
      v2f bfrag;
      bfrag.x = ldsB[kb * LDSW + j * 16 + l16];
      bfrag.y = ldsB[(kb + 1) * LDSW + j * 16 + l16];
      c = __builtin_amdgcn_wmma_f32_16x16x4_f32(
              /*neg_a=*/false, a[k], /*neg_b=*/false, bfrag,
              /*c_mod=*/(short)0, c, /*reuse_a=*/false, /*reuse_b=*/false);
    }
#pragma unroll
    for (int g = 0; g < 8; ++g) {
      if (valid[g]) {
        const float s2 = s2row[g][j * 16 + l16];
        acc += __expf(fabsf(c[g] - s2));
      }
    }
  }

  // wave32 reduction, one atomic per wave
#pragma unroll
  for (int off = 16; off >= 1; off >>= 1)
    acc += __shfl_xor(acc, off, 32);
  if (lane == 0) atomicAdd(per_expert + n, acc);
}

__global__ void mekoo_zero(float* per_expert) {
  if (threadIdx.x < NK_) per_expert[threadIdx.x] = 0.0f;
}

__global__ void mekoo_final(const float* __restrict__ Kp,
                            const float* __restrict__ per_expert,
                            float* __restrict__ out) {
  __shared__ float pe[NK_];
  if (threadIdx.x < NK_) pe[threadIdx.x] = per_expert[threadIdx.x];
  __syncthreads();
  float tot = 0.0f;
#pragma unroll
  for (int k = 0; k < NK_; ++k) tot = tot + pe[k];
  const int idx = blockIdx.x * blockDim.x + threadIdx.x;
  if (idx < D_ * D_) {
    float s = 0.0f;
#pragma unroll
    for (int k = 0; k < NK_; ++k)
      s += Kp[(size_t)k * D_ * D_ + idx] * (pe[k] / tot);
    out[idx] = s;
  }
}

extern "C" void kernel_launch(void* const* d_in, const int* in_sizes, int n_in,
                              void* d_out, int out_size, void* d_ws, size_t ws_size,
                              hipStream_t stream) {
  const float* seq = (const float*)d_in[0];   // (8, 2048, 128) f32
  const float* Kp  = (const float*)d_in[1];   // (16, 128, 128) f32
  float* out = (float*)d_out;                 // (128, 128) f32
  float* pe  = (float*)d_ws;                  // 16 f32 accumulators

  mekoo_zero<<<1, 32, 0, stream>>>(pe);
  dim3 grid(NTILES / WPB, NK_);
  mekoo_main<<<grid, 32 * WPB, 0, stream>>>(seq, Kp, pe);
  mekoo_final<<<(D_ * D_ + 255) / 256, 256, 0, stream>>>(Kp, pe, out);
}